// SeparableConv3D_28750511079631
// MI455X (gfx1250) — compile-verified
//
#include <hip/hip_runtime.h>
#include <cstdint>
#include <cstddef>

// Problem constants (match reference)
#define Nn 8
#define Cc 3
#define Tt 32
#define Hh 256
#define Ww 256
#define Kk 5

// Tile: 32 (H) x 64 (W) outputs per workgroup, full T walked in-loop.
#define HT 32
#define WT 64
#define LH (HT + 4)    // 36 rows with halo
#define LW (WT + 4)    // 68 cols with halo
#define NTHREADS 256

#if defined(__HIP_DEVICE_COMPILE__) && defined(__gfx1250__) && \
    __has_builtin(__builtin_amdgcn_global_load_async_to_lds_b32)
#define HAVE_ASYNC 1
#else
#define HAVE_ASYNC 0
#endif

__device__ __forceinline__ void wait_async0() {
#if HAVE_ASYNC
#if __has_builtin(__builtin_amdgcn_s_wait_asynccnt)
  __builtin_amdgcn_s_wait_asynccnt(0);
#else
  asm volatile("s_wait_asynccnt 0" ::: "memory");
#endif
#endif
}

// One 4-byte element: async DMA global->LDS if available, else plain copy.
__device__ __forceinline__ void gl2lds_b32(const float* __restrict__ g, float* l) {
#if HAVE_ASYNC
  __builtin_amdgcn_global_load_async_to_lds_b32(
      (__attribute__((address_space(1))) int*)(const_cast<float*>(g)),
      (__attribute__((address_space(3))) int*)(l),
      /*imm offset*/ 0, /*cpol*/ 0);
#else
  *l = *g;
#endif
}

// Load one t-plane tile (LH x LW, zero-padded outside [0,H)x[0,W)) into LDS.
__device__ __forceinline__ void load_plane(const float* __restrict__ xplane,
                                           float* __restrict__ lbuf,
                                           int h0, int w0, int tid) {
  for (int idx = tid; idx < LH * LW; idx += NTHREADS) {
    const int row = idx / LW;
    const int col = idx - row * LW;
    const int gh = h0 + row;
    const int gw = w0 + col;
    float* dst = &lbuf[idx];
    if ((unsigned)gh < (unsigned)Hh && (unsigned)gw < (unsigned)Ww) {
      gl2lds_b32(xplane + (size_t)gh * Ww + gw, dst);
    } else {
      *dst = 0.0f;  // zero 'same' padding
    }
  }
}

__global__ __launch_bounds__(NTHREADS)
void sepconv3d_fused_kernel(const float* __restrict__ x,
                            const float* __restrict__ w1,   // [C][5] along W
                            const float* __restrict__ w2,   // [C][5] along H
                            const float* __restrict__ w3,   // [C][5] along T
                            float* __restrict__ out) {
  __shared__ float lin[2][LH * LW];   // double-buffered haloed input plane
  __shared__ float ltmp[LH * WT];     // after W-conv (rows still haloed in H)

  const int tid = threadIdx.x;
  const int bx = blockIdx.x;          // W tile (0..3)
  const int by = blockIdx.y;          // H tile (0..7)
  const int nc = blockIdx.z;          // n*C + c (0..23)
  const int c  = nc % Cc;

  const int h0 = by * HT - 2;
  const int w0 = bx * WT - 2;

  const float* __restrict__ xnc = x   + (size_t)nc * Tt * Hh * Ww;
  float* __restrict__       onc = out + (size_t)nc * Tt * Hh * Ww;

  float w1c[5], w2c[5], w3c[5];
#pragma unroll
  for (int i = 0; i < 5; ++i) {
    w1c[i] = w1[c * Kk + i];
    w2c[i] = w2[c * Kk + i];
    w3c[i] = w3[c * Kk + i];
  }

  const int tx = tid & (WT - 1);      // 0..63  -> output w offset
  const int ty = tid >> 6;            // 0..3   -> base output h row (stride 4)

  // Per-thread register ring over t: ring[k][jj] = r_{t-4+k} at row ty+4*jj.
  float ring[5][8];
#pragma unroll
  for (int k = 0; k < 5; ++k)
#pragma unroll
    for (int j = 0; j < 8; ++j) ring[k][j] = 0.0f;

  // Preload plane t=0 into buffer 0.
  load_plane(xnc, lin[0], h0, w0, tid);

  for (int t = 0; t < Tt + 2; ++t) {
    float rnew[8];
    if (t < Tt) {
      const int cur = t & 1;
      wait_async0();        // this wave's async writes for plane t complete
      __syncthreads();      // plane t visible WG-wide; ltmp & lin[cur^1] reusable

      // Prefetch plane t+1 into the other buffer while we compute plane t.
      if (t + 1 < Tt)
        load_plane(xnc + (size_t)(t + 1) * Hh * Ww, lin[cur ^ 1], h0, w0, tid);

      // Stage 1: W-conv  ltmp[row][w] = sum_i w1[i] * lin[row][w+i]
      // Exactly (LH*WT)/NTHREADS == 9 iterations: force static unroll.
      const float* __restrict__ li = lin[cur];
#pragma unroll
      for (int it = 0; it < (LH * WT) / NTHREADS; ++it) {
        const int idx = it * NTHREADS + tid;
        const int row = idx >> 6;       // 0..35
        const int col = idx & (WT - 1); // 0..63
        const float* p = &li[row * LW + col];
        float a = 0.0f;
#pragma unroll
        for (int i = 0; i < 5; ++i) a += w1c[i] * p[i];
        ltmp[idx] = a;
      }
      __syncthreads();

      // Stage 2: H-conv -> registers   rnew[jj] at row (ty + 4*jj)
#pragma unroll
      for (int jj = 0; jj < 8; ++jj) {
        const float* q = &ltmp[(ty + 4 * jj) * WT + tx];
        float a = 0.0f;
#pragma unroll
        for (int j = 0; j < 5; ++j) a += w2c[j] * q[j * WT];
        rnew[jj] = a;
      }
    } else {
      // Virtual zero planes past the end (T-axis zero padding tail).
#pragma unroll
      for (int jj = 0; jj < 8; ++jj) rnew[jj] = 0.0f;
    }

    // Stage 3: push into T ring (pure register moves).
#pragma unroll
    for (int k = 0; k < 4; ++k)
#pragma unroll
      for (int jj = 0; jj < 8; ++jj) ring[k][jj] = ring[k + 1][jj];
#pragma unroll
    for (int jj = 0; jj < 8; ++jj) ring[4][jj] = rnew[jj];

    // Emit output plane to = t-2 (needs r_{to-2..to+2} = ring[0..4]).
    // Non-temporal stores: output is write-once, keep it out of L2 so the
    // 192MB L2 holds input halo lines that neighboring workgroups re-read.
    const int to = t - 2;
    if (to >= 0) {
#pragma unroll
      for (int jj = 0; jj < 8; ++jj) {
        float a = 0.0f;
#pragma unroll
        for (int k = 0; k < 5; ++k) a += w3c[k] * ring[k][jj];
        const int gh = by * HT + ty + 4 * jj;
        __builtin_nontemporal_store(
            a, &onc[((size_t)to * Hh + gh) * Ww + (bx * WT + tx)]);
      }
    }
  }
}

extern "C" void kernel_launch(void* const* d_in, const int* in_sizes, int n_in,
                              void* d_out, int out_size, void* d_ws, size_t ws_size,
                              hipStream_t stream) {
  const float* x  = (const float*)d_in[0];
  const float* w1 = (const float*)d_in[1];
  const float* w2 = (const float*)d_in[2];
  const float* w3 = (const float*)d_in[3];
  float* out = (float*)d_out;

  dim3 grid(Ww / WT, Hh / HT, Nn * Cc);  // 4 x 8 x 24 = 768 workgroups
  dim3 block(NTHREADS);                  // 8 waves (wave32)
  sepconv3d_fused_kernel<<<grid, block, 0, stream>>>(x, w1, w2, w3, out);
}